// Aggregator_35837207118174
// MI455X (gfx1250) — compile-verified
//
#include <hip/hip_runtime.h>
#include <hip/hip_bf16.h>

#define N_USERS    50000
#define N_ITEMS    30000
#define N_ENTITIES 100000
#define N_RELATIONS 50
#define N_EDGES    1500000
#define N_INTER    1000000
#define DIM        64
#define GAMMA      0.1f

typedef __attribute__((ext_vector_type(2))) float v2f;
typedef __attribute__((ext_vector_type(8))) float v8f;

__device__ __forceinline__ float sigmoidf(float x) { return 1.0f / (1.0f + __expf(-x)); }

// ---------------------------------------------------------------------------
// Stage A: scatter KG edges into per-head sums + counts
// thread = (edge, dim)
// ---------------------------------------------------------------------------
__global__ __launch_bounds__(256) void edge_scatter_kernel(
    const int* __restrict__ ei, const int* __restrict__ et,
    const float* __restrict__ ent, const float* __restrict__ relw,
    float* __restrict__ sum1, float* __restrict__ sum2,
    float* __restrict__ cntc, float* __restrict__ cnts,
    float* __restrict__ relsum)
{
    long long gid = (long long)blockIdx.x * blockDim.x + threadIdx.x;
    if (gid >= (long long)N_EDGES * DIM) return;
    int e = (int)(gid >> 6);
    int d = (int)(gid & 63);
    int head = ei[e];
    int tail = ei[N_EDGES + e];
    int r    = et[e];
    float te = ent[(long long)tail * DIM + d];
    float re = relw[r * DIM + d];
    bool cross = (head < N_ITEMS) != (tail < N_ITEMS);
    if (cross) atomicAdd(&sum1[(long long)head * DIM + d], te * re);
    else       atomicAdd(&sum2[(long long)head * DIM + d], te + re);
    if (head < N_ITEMS) atomicAdd(&relsum[head * DIM + d], re);
    if (d == 0) atomicAdd(cross ? &cntc[head] : &cnts[head], 1.0f);
}

// ---------------------------------------------------------------------------
// Stage B: entity MLP via fp32 WMMA.
// out = lrelu(mean1 @ W1^T + b1)/2 + lrelu(mean2 @ W2^T + b2)/2
// One wave handles a 16-row entity tile x all 64 output cols (4 N-tiles).
// A fragment (16x4 f32, 2 VGPRs): lane<16 -> K=k0,k0+1 ; lane>=16 -> K=k0+2,k0+3
// B fragment (4x16 f32, 2 VGPRs): mirrored K split, N = lane&15
// D (16x16 f32, 8 VGPRs): elem v -> M = v + 8*(lane>=16), N = lane&15
// ---------------------------------------------------------------------------
__global__ __launch_bounds__(256) void entity_mlp_wmma_kernel(
    const float* __restrict__ sum1, const float* __restrict__ sum2,
    const float* __restrict__ cntc, const float* __restrict__ cnts,
    const float* __restrict__ W1, const float* __restrict__ bias1,
    const float* __restrict__ W2, const float* __restrict__ bias2,
    float* __restrict__ out)
{
    __shared__ float lw1[64 * 66];
    __shared__ float lw2[64 * 66];
    for (int i = threadIdx.x; i < 64 * 64; i += 256) {
        int n = i >> 6, k = i & 63;
        lw1[n * 66 + k] = W1[i];
        lw2[n * 66 + k] = W2[i];
    }
    __syncthreads();

    const int wave = threadIdx.x >> 5;
    const int lane = threadIdx.x & 31;
    const int tile = blockIdx.x * 8 + wave;
    if (tile >= (N_ENTITIES + 15) / 16) return;   // wave-uniform branch: EXEC stays all-ones

    const int m_base = tile * 16;
    const int hi = lane >> 4;        // 0: K pair {0,1}, 1: K pair {2,3}
    const int mr = lane & 15;
    const int row = m_base + mr;

    const float rc1 = 1.0f / fmaxf(cntc[row], 1.0f);
    const float rc2 = 1.0f / fmaxf(cnts[row], 1.0f);
    const float* s1r = sum1 + (long long)row * DIM;
    const float* s2r = sum2 + (long long)row * DIM;

    v8f acc1[4], acc2[4];
#pragma unroll
    for (int t = 0; t < 4; t++) {
        acc1[t] = (v8f){0.f,0.f,0.f,0.f,0.f,0.f,0.f,0.f};
        acc2[t] = (v8f){0.f,0.f,0.f,0.f,0.f,0.f,0.f,0.f};
    }

#pragma unroll 4
    for (int k0 = 0; k0 < 64; k0 += 4) {
        const int ka = k0 + 2 * hi;
        v2f a1, a2;
        a1.x = s1r[ka] * rc1;  a1.y = s1r[ka + 1] * rc1;
        a2.x = s2r[ka] * rc2;  a2.y = s2r[ka + 1] * rc2;
#pragma unroll
        for (int t = 0; t < 4; t++) {
            const int n = t * 16 + mr;
            v2f bv1, bv2;
            bv1.x = lw1[n * 66 + ka];  bv1.y = lw1[n * 66 + ka + 1];
            bv2.x = lw2[n * 66 + ka];  bv2.y = lw2[n * 66 + ka + 1];
            acc1[t] = __builtin_amdgcn_wmma_f32_16x16x4_f32(
                false, a1, false, bv1, (short)0, acc1[t], false, false);
            acc2[t] = __builtin_amdgcn_wmma_f32_16x16x4_f32(
                false, a2, false, bv2, (short)0, acc2[t], false, false);
        }
    }

#pragma unroll
    for (int t = 0; t < 4; t++) {
        const int n = t * 16 + mr;
        const float bb1 = bias1[n];
        const float bb2 = bias2[n];
#pragma unroll
        for (int v = 0; v < 8; v++) {
            const int m = m_base + v + 8 * hi;
            float x1 = acc1[t][v] + bb1;
            x1 = (x1 >= 0.f ? x1 : 0.01f * x1) * 0.5f;
            float x2 = acc2[t][v] + bb2;
            x2 = (x2 >= 0.f ? x2 : 0.01f * x2) * 0.5f;
            out[(long long)m * DIM + n] = x1 + x2;
        }
    }
}

// ---------------------------------------------------------------------------
// w_item[i,d] = (relsum[i,d] / max(cnt_total[i],1)) * entity_emb[i,d]
// ---------------------------------------------------------------------------
__global__ __launch_bounds__(256) void make_witem_kernel(
    const float* __restrict__ relsum, const float* __restrict__ cntc,
    const float* __restrict__ cnts, const float* __restrict__ ent,
    float* __restrict__ w_item)
{
    int i = blockIdx.x * blockDim.x + threadIdx.x;
    if (i >= N_ITEMS * DIM) return;
    int item = i >> 6;
    float c = fmaxf(cntc[item] + cnts[item], 1.0f);
    w_item[i] = (relsum[i] / c) * ent[i];
}

// ---------------------------------------------------------------------------
// per-interaction dots -> sigmoid -> segment max (int-punned atomicMax; all
// values in (0,1) so positive-float ordering == int ordering)
// ---------------------------------------------------------------------------
__global__ __launch_bounds__(256) void dot_sig_max_kernel(
    const int* __restrict__ im,
    const float* __restrict__ ue, const float* __restrict__ uce,
    const float* __restrict__ w_item, const float* __restrict__ icf,
    float* __restrict__ p, float* __restrict__ pcf,
    float* __restrict__ smax_p, float* __restrict__ smax_pcf)
{
    int e = blockIdx.x * blockDim.x + threadIdx.x;
    if (e >= N_INTER) return;
    int row = im[2 * e], col = im[2 * e + 1];
    const float4* a  = (const float4*)(ue     + (long long)row * DIM);
    const float4* b  = (const float4*)(w_item + (long long)col * DIM);
    const float4* a2 = (const float4*)(uce    + (long long)row * DIM);
    const float4* b2 = (const float4*)(icf    + (long long)col * DIM);
    float s = 0.f, s2 = 0.f;
#pragma unroll
    for (int k = 0; k < 16; k++) {
        float4 va = a[k],  vb = b[k];
        s  += va.x * vb.x + va.y * vb.y + va.z * vb.z + va.w * vb.w;
        float4 vc = a2[k], vd = b2[k];
        s2 += vc.x * vd.x + vc.y * vd.y + vc.z * vd.z + vc.w * vd.w;
    }
    float pv = sigmoidf(s), pcfv = sigmoidf(s2);
    p[e] = pv;  pcf[e] = pcfv;
    atomicMax((int*)&smax_p[row],   __float_as_int(pv));
    atomicMax((int*)&smax_pcf[row], __float_as_int(pcfv));
}

__global__ __launch_bounds__(256) void exp_sum_kernel(
    const int* __restrict__ im,
    float* __restrict__ p, float* __restrict__ pcf,
    const float* __restrict__ smax_p, const float* __restrict__ smax_pcf,
    float* __restrict__ ssum_p, float* __restrict__ ssum_pcf)
{
    int e = blockIdx.x * blockDim.x + threadIdx.x;
    if (e >= N_INTER) return;
    int row = im[2 * e];
    float ep = __expf(p[e]   - smax_p[row]);
    float ec = __expf(pcf[e] - smax_pcf[row]);
    p[e] = ep;  pcf[e] = ec;
    atomicAdd(&ssum_p[row], ep);
    atomicAdd(&ssum_pcf[row], ec);
}

// thread = (interaction, dim): masked weighted scatter into u / ucf
__global__ __launch_bounds__(256) void accum_kernel(
    const int* __restrict__ im,
    const float* __restrict__ p, const float* __restrict__ pcf,
    const float* __restrict__ ssum_p, const float* __restrict__ ssum_pcf,
    const float* __restrict__ item_kg, const float* __restrict__ icf,
    float* __restrict__ u, float* __restrict__ ucf,
    float* __restrict__ mask_out, int write_mask)
{
    long long gid = (long long)blockIdx.x * blockDim.x + threadIdx.x;
    if (gid >= (long long)N_INTER * DIM) return;
    int e = (int)(gid >> 6);
    int d = (int)(gid & 63);
    int row = im[2 * e], col = im[2 * e + 1];
    float ps  = p[e]   / ssum_p[row];
    float pcs = pcf[e] / ssum_pcf[row];
    float m = (fabsf(sigmoidf(ps) - sigmoidf(pcs)) < GAMMA) ? 1.0f : 0.0f;
    atomicAdd(&u[(long long)row * DIM + d],   item_kg[(long long)col * DIM + d] * ps  * m);
    atomicAdd(&ucf[(long long)row * DIM + d], icf[(long long)col * DIM + d]     * pcs * m);
    if (write_mask && d == 0) mask_out[e] = m;
}

// one wave per user row; L2-normalize u and ucf in place
__global__ __launch_bounds__(256) void normalize_rows_kernel(
    float* __restrict__ u, float* __restrict__ ucf)
{
    int wave = threadIdx.x >> 5, lane = threadIdx.x & 31;
    int row = blockIdx.x * 8 + wave;
    if (row >= N_USERS) return;
    float* pr = u + (long long)row * DIM;
    float a = pr[lane], b = pr[lane + 32];
    float s = a * a + b * b;
#pragma unroll
    for (int o = 16; o >= 1; o >>= 1) s += __shfl_xor(s, o, 32);
    float inv = 1.0f / fmaxf(sqrtf(s), 1e-12f);
    pr[lane] = a * inv;  pr[lane + 32] = b * inv;

    float* pc = ucf + (long long)row * DIM;
    a = pc[lane];  b = pc[lane + 32];
    s = a * a + b * b;
#pragma unroll
    for (int o = 16; o >= 1; o >>= 1) s += __shfl_xor(s, o, 32);
    inv = 1.0f / fmaxf(sqrtf(s), 1e-12f);
    pc[lane] = a * inv;  pc[lane + 32] = b * inv;
}

// item_agg scatter-mean
__global__ __launch_bounds__(256) void item_scatter_kernel(
    const int* __restrict__ im, const float* __restrict__ user_cf,
    float* __restrict__ isum, float* __restrict__ icnt)
{
    long long gid = (long long)blockIdx.x * blockDim.x + threadIdx.x;
    if (gid >= (long long)N_INTER * DIM) return;
    int e = (int)(gid >> 6);
    int d = (int)(gid & 63);
    int row = im[2 * e], col = im[2 * e + 1];
    atomicAdd(&isum[(long long)col * DIM + d], user_cf[(long long)row * DIM + d]);
    if (d == 0) atomicAdd(&icnt[col], 1.0f);
}

__global__ __launch_bounds__(256) void item_div_kernel(
    float* __restrict__ isum, const float* __restrict__ icnt)
{
    int i = blockIdx.x * blockDim.x + threadIdx.x;
    if (i >= N_ITEMS * DIM) return;
    isum[i] /= fmaxf(icnt[i >> 6], 1.0f);
}

// ---------------------------------------------------------------------------
extern "C" void kernel_launch(void* const* d_in, const int* in_sizes, int n_in,
                              void* d_out, int out_size, void* d_ws, size_t ws_size,
                              hipStream_t stream)
{
    const float* ent  = (const float*)d_in[0];
    const float* uemb = (const float*)d_in[1];
    const float* ucf0 = (const float*)d_in[2];
    const float* icf  = (const float*)d_in[3];
    const float* relw = (const float*)d_in[4];
    const float* W1   = (const float*)d_in[5];
    const float* B1   = (const float*)d_in[6];
    const float* W2   = (const float*)d_in[7];
    const float* B2   = (const float*)d_in[8];
    const int*   ei   = (const int*)d_in[9];
    const int*   et   = (const int*)d_in[10];
    const int*   im   = (const int*)d_in[11];

    float* out = (float*)d_out;
    float* out_ent  = out;                                        // 100000*64
    float* out_u    = out + (size_t)N_ENTITIES * DIM;             // 50000*64
    float* out_ucf  = out_u + (size_t)N_USERS * DIM;              // 50000*64
    float* out_item = out_ucf + (size_t)N_USERS * DIM;            // 30000*64
    float* out_mask = out_item + (size_t)N_ITEMS * DIM;           // 1000000

    float* ws = (float*)d_ws;
    size_t off = 0;
    float* sum1     = ws + off; off += (size_t)N_ENTITIES * DIM;
    float* sum2     = ws + off; off += (size_t)N_ENTITIES * DIM;
    float* cntc     = ws + off; off += N_ENTITIES;
    float* cnts     = ws + off; off += N_ENTITIES;
    float* relsum   = ws + off; off += (size_t)N_ITEMS * DIM;
    float* w_item   = ws + off; off += (size_t)N_ITEMS * DIM;
    float* pbuf     = ws + off; off += N_INTER;
    float* pcfbuf   = ws + off; off += N_INTER;
    float* smax_p   = ws + off; off += N_USERS;
    float* smax_pcf = ws + off; off += N_USERS;
    float* ssum_p   = ws + off; off += N_USERS;
    float* ssum_pcf = ws + off; off += N_USERS;
    float* icnt     = ws + off; off += N_ITEMS;

    // ---- Stage A: KG edge aggregation (sum1..relsum are contiguous) ----
    size_t stageA_floats = (size_t)N_ENTITIES * DIM * 2 + 2 * N_ENTITIES + (size_t)N_ITEMS * DIM;
    hipMemsetAsync(sum1, 0, stageA_floats * sizeof(float), stream);
    {
        long long n = (long long)N_EDGES * DIM;
        edge_scatter_kernel<<<(unsigned)((n + 255) / 256), 256, 0, stream>>>(
            ei, et, ent, relw, sum1, sum2, cntc, cnts, relsum);
    }

    // ---- Stage B: entity MLP via fp32 WMMA ----
    {
        int tiles = (N_ENTITIES + 15) / 16;       // 6250
        entity_mlp_wmma_kernel<<<(tiles + 7) / 8, 256, 0, stream>>>(
            sum1, sum2, cntc, cnts, W1, B1, W2, B2, out_ent);
    }

    // ---- w_item = rel_i * item_kg precompute ----
    make_witem_kernel<<<(N_ITEMS * DIM + 255) / 256, 256, 0, stream>>>(
        relsum, cntc, cnts, ent, w_item);

    // ---- item_agg (independent of iterations) ----
    hipMemsetAsync(out_item, 0, (size_t)N_ITEMS * DIM * sizeof(float), stream);
    hipMemsetAsync(icnt, 0, N_ITEMS * sizeof(float), stream);
    {
        long long n = (long long)N_INTER * DIM;
        item_scatter_kernel<<<(unsigned)((n + 255) / 256), 256, 0, stream>>>(
            im, ucf0, out_item, icnt);
    }
    item_div_kernel<<<(N_ITEMS * DIM + 255) / 256, 256, 0, stream>>>(out_item, icnt);

    // ---- Stage C: 3 softmax-attention iterations ----
    for (int it = 0; it < 3; it++) {
        const float* ue  = (it == 0) ? uemb : out_u;
        const float* uce = (it == 0) ? ucf0 : out_ucf;

        // smax_p, smax_pcf, ssum_p, ssum_pcf are contiguous
        hipMemsetAsync(smax_p, 0, (size_t)4 * N_USERS * sizeof(float), stream);

        dot_sig_max_kernel<<<(N_INTER + 255) / 256, 256, 0, stream>>>(
            im, ue, uce, w_item, icf, pbuf, pcfbuf, smax_p, smax_pcf);

        // old u/ucf fully consumed by dot kernel above; zero accumulators now
        hipMemsetAsync(out_u, 0, (size_t)2 * N_USERS * DIM * sizeof(float), stream);

        exp_sum_kernel<<<(N_INTER + 255) / 256, 256, 0, stream>>>(
            im, pbuf, pcfbuf, smax_p, smax_pcf, ssum_p, ssum_pcf);

        {
            long long n = (long long)N_INTER * DIM;
            accum_kernel<<<(unsigned)((n + 255) / 256), 256, 0, stream>>>(
                im, pbuf, pcfbuf, ssum_p, ssum_pcf, ent /* item_kg = entity_emb[:N_ITEMS] */,
                icf, out_u, out_ucf, out_mask, (it == 2) ? 1 : 0);
        }

        if (it < 2) {
            normalize_rows_kernel<<<(N_USERS + 7) / 8, 256, 0, stream>>>(out_u, out_ucf);
        }
    }

    (void)in_sizes; (void)n_in; (void)out_size; (void)ws_size;
}